// RNN_86818468921879
// MI455X (gfx1250) — compile-verified
//
#include <hip/hip_runtime.h>
#include <hip/hip_bf16.h>
#include <math.h>

// ---------------------------------------------------------------------------
// RNN on MI455X (gfx1250), fp32 throughout via V_WMMA_F32_16X16X4_F32.
//
// Phase 1: xp = x @ Wx + b  written straight into d_out (N,T,H).
// Phase 2: persistent recurrence kernel, 8 batch-groups x 16 col-blocks,
//          in-place overwrite of xp with h, per-group atomic barrier per step.
//
// A tiles (16x1024 fp32) are DMA'd into LDS by the Tensor Data Mover
// (tensor_load_to_lds, TENSORcnt) in iterate mode with an LDS row stride of
// 1028 dwords: fragment reads are conflict-free ds_load_b64 with pure
// immediate offsets (no per-iteration address VALU). A VALU staging fallback
// produces the identical layout when the TDM builtin is unavailable.
// B matrices are pre-packed in fragment order (256 B/wave contiguous bursts).
//
// Sizes (fixed by the reference): N=128, T=512, D=H=1024.
// ---------------------------------------------------------------------------

#define RNN_N 128
#define RNN_T 512
#define RNN_D 1024
#define RNN_H 1024

#define LDS_ROW_STRIDE 1028   // dwords; bank = (4*row + 4*kc + 2*half) -> conflict-free

typedef __attribute__((ext_vector_type(2))) float v2f;
typedef __attribute__((ext_vector_type(8))) float v8f;
typedef __attribute__((ext_vector_type(4))) unsigned u32x4;
typedef __attribute__((ext_vector_type(4))) int i32x4;
typedef __attribute__((ext_vector_type(8))) int i32x8;

#if defined(__has_builtin)
#  if __has_builtin(__builtin_amdgcn_tensor_load_to_lds) && \
      __has_builtin(__builtin_amdgcn_s_wait_tensorcnt)
#    define RNN_USE_TDM 1
#  endif
#endif
#ifndef RNN_USE_TDM
#  define RNN_USE_TDM 0
#endif

// ---------------------------------------------------------------------------
// Fragment layout for V_WMMA_F32_16X16X4_F32 (wave32):
//   A (16x4, MxK): lane l (0-15): row M=l, regs = K{0,1}; lanes 16-31: K{2,3}.
//   B (4x16, KxN): mirrored: lane l: col N=l%16, regs = K{(l/16)*2, +1}.
//   C/D (16x16):   reg r, lanes 0-15 -> (M=r, N=lane); lanes 16-31 -> (M=r+8).
//
// Packed B:   pack[((ct*256 + kc)*32 + lane)*2 + j]
//                 = W[(kc*4 + (lane>>4)*2 + j)*H + ct*16 + (lane&15)]
// LDS A tile: Ash[row*1028 + k]  (row stride 1028 dwords, TDM iterate mode)
// ---------------------------------------------------------------------------

__global__ void rnn_pack_b_kernel(const float* __restrict__ W,
                                  float* __restrict__ pack) {
    int idx  = blockIdx.x * blockDim.x + threadIdx.x;   // [0, 64*256*32)
    int lane = idx & 31;
    int kc   = (idx >> 5) & 255;
    int ct   = idx >> 13;                               // col tile [0,64)
    int col  = ct * 16 + (lane & 15);
    int k    = kc * 4 + ((lane >> 4) << 1);
    v2f v;
    v[0] = W[(size_t)k * RNN_H + col];
    v[1] = W[(size_t)(k + 1) * RNN_H + col];
    *(v2f*)(pack + (size_t)idx * 2) = v;
}

__device__ __forceinline__ v8f wmma_f32_k4(v2f a, v2f b, v8f c) {
    return __builtin_amdgcn_wmma_f32_16x16x4_f32(
        /*neg_a=*/false, a, /*neg_b=*/false, b,
        /*c_mod=*/(short)0, c, /*reuse_a=*/false, /*reuse_b=*/false);
}

#if RNN_USE_TDM
// ---------------------------------------------------------------------------
// TDM: DMA one 16x1024 fp32 tile (global row stride `strideE` elements) into
// LDS at `lds_byte_addr`, one row per iteration, LDS rows 1028 dwords apart.
// D# bit layout per CDNA5 ISA ch.8 (groups 0..3); issued by one wave, tracked
// on TENSORcnt.
// ---------------------------------------------------------------------------
__device__ __forceinline__ void tdm_load_a_tile(const float* src, size_t strideE,
                                                unsigned lds_byte_addr) {
    unsigned long long ga = (unsigned long long)(const void*)src;
    u32x4 g0;
    g0[0] = 1u;                                          // count=1, user mode
    g0[1] = lds_byte_addr;                               // lds_addr [63:32]
    g0[2] = (unsigned)ga;                                // global_addr [95:64]
    g0[3] = (unsigned)((ga >> 32) & 0x1FFFFFFull)        // global_addr [120:96]
            | (2u << 30);                                // type=2 ("image")
    i32x8 g1;
    g1[0] = (int)((2u << 16) | (1u << 19));   // data_size=4B, iterate_enable=1
    g1[1] = (int)(1024u << 16);               // tensor_dim0[15:0]=1024
    g1[2] = (int)(1u << 16);                  // tensor_dim0[31:16]=0 | tensor_dim1=1
    g1[3] = (int)(1024u << 16);               // tensor_dim1 hi=0 | tile_dim0=1024
    g1[4] = 1;                                // tile_dim1=1, tile_dim2=0
    g1[5] = (int)(unsigned)strideE;           // tensor_dim0_stride[31:0]
    g1[6] = (int)((strideE >> 32) & 0xFFFF);  // tensor_dim0_stride[47:32]
    g1[7] = 0;
    i32x4 g2;
    g2[0] = 0;                                // tensor_dim2 (unused)
    g2[1] = LDS_ROW_STRIDE;                   // lds_addr_increment per iteration
    g2[2] = (int)(unsigned)strideE;           // global_addr_increment[31:0]
    g2[3] = (int)((15u << 16)                 // iterate_count=15 -> 16 rows
            | (unsigned)((strideE >> 32) & 0xFFFF));
    i32x4 g3 = {0, 0, 0, 0};
#if __clang_major__ >= 23
    i32x8 g4 = {0, 0, 0, 0, 0, 0, 0, 0};
    __builtin_amdgcn_tensor_load_to_lds(g0, g1, g2, g3, g4, 0);
#else
    __builtin_amdgcn_tensor_load_to_lds(g0, g1, g2, g3, 0);
#endif
}
#endif  // RNN_USE_TDM

// VALU fallback staging: identical LDS layout (row stride 1028 dwords).
template <int NTHREADS>
__device__ __forceinline__ void stage_a_tile_valu(const float* __restrict__ Ab,
                                                  size_t strideE, float* Ash,
                                                  int tid) {
#pragma unroll
    for (int i = 0; i < 4096 / NTHREADS; ++i) {
        int f   = tid + i * NTHREADS;      // float4 index over the 16x1024 tile
        int row = f >> 8;                  // [0,16)
        int kv  = f & 255;
        const float4 a = *(const float4*)(Ab + (size_t)row * strideE
                                             + (size_t)kv * 4);
        *(float4*)&Ash[row * LDS_ROW_STRIDE + kv * 4] = a;
    }
}

// C(16x16) += A_lds(16x1024) * packedB(colTile); 4 accumulator chains.
// All A addresses are base + immediate (kc*16 bytes): zero address VALU.
__device__ __forceinline__ v8f gemm_k1024_lds(const float* Ash, int lane,
                                              const float* __restrict__ bbase) {
    v8f c0 = {}, c1 = {}, c2 = {}, c3 = {};
    const float* abase = Ash + (lane & 15) * LDS_ROW_STRIDE + ((lane >> 4) << 1);
#pragma unroll 4
    for (int kc = 0; kc < 256; kc += 4) {
        v2f a0 = *(const v2f*)(abase + (kc + 0) * 4);
        v2f a1 = *(const v2f*)(abase + (kc + 1) * 4);
        v2f a2 = *(const v2f*)(abase + (kc + 2) * 4);
        v2f a3 = *(const v2f*)(abase + (kc + 3) * 4);
        v2f b0 = *(const v2f*)(bbase + (size_t)(kc + 0) * 64);
        v2f b1 = *(const v2f*)(bbase + (size_t)(kc + 1) * 64);
        v2f b2 = *(const v2f*)(bbase + (size_t)(kc + 2) * 64);
        v2f b3 = *(const v2f*)(bbase + (size_t)(kc + 3) * 64);
        c0 = wmma_f32_k4(a0, b0, c0);
        c1 = wmma_f32_k4(a1, b1, c1);
        c2 = wmma_f32_k4(a2, b2, c2);
        c3 = wmma_f32_k4(a3, b3, c3);
    }
    return (c0 + c1) + (c2 + c3);
}

// ---------------------------------------------------------------------------
// Phase 1: xp = x @ Wx + b -> d_out.  Block = 256 thr = 8 waves, one shared
// 16-row A tile x 8 col-tiles (16 rows x 128 cols per block).
// Grid = (65536/16, 1024/128) = (4096, 8).
// ---------------------------------------------------------------------------
__global__ void rnn_xproj_kernel(const float* __restrict__ x,
                                 const float* __restrict__ packWx,
                                 const float* __restrict__ bias,
                                 float* __restrict__ out) {
    __shared__ float Ash[16 * LDS_ROW_STRIDE];        // 64.25 KB
    const int tid  = threadIdx.x;
    const int lane = tid & 31;
    const int wave = tid >> 5;                        // 0..7
    const int rowTile = blockIdx.x;                   // [0,4096)
    const int colTile = blockIdx.y * 8 + wave;        // [0,64)
    const float* Ab = x + (size_t)rowTile * 16 * RNN_D;

#if RNN_USE_TDM
    if (wave == 0) {
        tdm_load_a_tile(Ab, RNN_D, (unsigned)(size_t)Ash);
        __builtin_amdgcn_s_wait_tensorcnt(0);
    }
#else
    stage_a_tile_valu<256>(Ab, RNN_D, Ash, tid);
#endif
    __syncthreads();

    const float* bbase = packWx + ((size_t)colTile * 256 * 32 + lane) * 2;
    v8f acc = gemm_k1024_lds(Ash, lane, bbase);

    const int col = colTile * 16 + (lane & 15);
    const float bv = bias[col];
    float* op = out + (size_t)rowTile * 16 * RNN_H + col;
#pragma unroll
    for (int r = 0; r < 8; ++r) {
        int m = r + ((lane >> 4) << 3);
        op[(size_t)m * RNN_H] = acc[r] + bv;
    }
}

// ---------------------------------------------------------------------------
// Phase 2: persistent recurrence. 128 blocks x 128 thr (4 waves).
//   group g = blockIdx.x>>4 owns batch rows [16g, 16g+16)  (M=16 tile)
//   the 16 blocks of a group cover all 1024 H-columns (4 col-tiles/block).
// Per step: TDM h_{t-1} (or h0) into LDS, WMMA against packed Wh, add xp
// (already in d_out at (n,t)), tanh, write back in place, then device-scope
// arrive/spin barrier among the group's 16 blocks.
// ---------------------------------------------------------------------------
__global__ void rnn_recur_kernel(const float* __restrict__ h0,
                                 const float* __restrict__ packWh,
                                 float* __restrict__ out,
                                 unsigned* __restrict__ ctr) {
    __shared__ float Ash[16 * LDS_ROW_STRIDE];        // 64.25 KB
    const int tid  = threadIdx.x;                     // 0..127
    const int lane = tid & 31;
    const int wave = tid >> 5;                        // 0..3
    const int g    = blockIdx.x >> 4;                 // batch group [0,8)
    const int cg   = blockIdx.x & 15;                 // column block [0,16)
    const int colTile = cg * 4 + wave;                // [0,64)
    const int col  = colTile * 16 + (lane & 15);
    const size_t TH = (size_t)RNN_T * RNN_H;
    const size_t rowbase = (size_t)g * 16;

    const float* bbase = packWh + ((size_t)colTile * 256 * 32 + lane) * 2;
    unsigned* myctr = ctr + g;

    for (int t = 0; t < RNN_T; ++t) {
        // xp for this step streams from HBM (written long ago by phase 1):
        // prefetch this block's 16x64 xp tile while the TDM runs.
        if (tid < 32) {
            const float* pf = out + (rowbase + (tid & 15)) * TH
                                  + (size_t)t * RNN_H + cg * 64 + ((tid >> 4) << 5);
            __builtin_prefetch(pf, 0, 0);
        }

        const float* Ab;
        size_t strideE;
        if (t == 0) { Ab = h0 + rowbase * RNN_H;                          strideE = RNN_H; }
        else        { Ab = out + rowbase * TH + (size_t)(t - 1) * RNN_H;  strideE = TH; }

#if RNN_USE_TDM
        if (wave == 0) {
            tdm_load_a_tile(Ab, strideE, (unsigned)(size_t)Ash);
            __builtin_amdgcn_s_wait_tensorcnt(0);
        }
#else
        stage_a_tile_valu<128>(Ab, strideE, Ash, tid);
#endif
        __syncthreads();

        v8f acc = gemm_k1024_lds(Ash, lane, bbase);

#pragma unroll
        for (int r = 0; r < 8; ++r) {
            int m = r + ((lane >> 4) << 3);
            float* p = out + (rowbase + m) * TH + (size_t)t * RNN_H + col;
            *p = tanhf(*p + acc[r]);
        }

        // ---- group barrier: all 16 blocks of group g must finish step t ----
        // (its two __syncthreads also protect Ash against early re-staging)
        __threadfence();                       // make h_t visible device-wide
        __syncthreads();
        if (tid == 0) {
            __hip_atomic_fetch_add(myctr, 1u, __ATOMIC_ACQ_REL,
                                   __HIP_MEMORY_SCOPE_AGENT);
            const unsigned want = 16u * (unsigned)(t + 1);
            while (__hip_atomic_load(myctr, __ATOMIC_ACQUIRE,
                                     __HIP_MEMORY_SCOPE_AGENT) < want) {
                __builtin_amdgcn_s_sleep(2);
            }
        }
        __syncthreads();
    }
}

// ---------------------------------------------------------------------------
// Launch. Inputs (setup_inputs order): x, h0, Wx, Wh, b.
// Workspace: [0,4MB) packed Wx frags, [4MB,8MB) packed Wh frags, [8MB) counters.
// ---------------------------------------------------------------------------
extern "C" void kernel_launch(void* const* d_in, const int* in_sizes, int n_in,
                              void* d_out, int out_size, void* d_ws, size_t ws_size,
                              hipStream_t stream) {
    const float* x  = (const float*)d_in[0];
    const float* h0 = (const float*)d_in[1];
    const float* Wx = (const float*)d_in[2];
    const float* Wh = (const float*)d_in[3];
    const float* b  = (const float*)d_in[4];
    float* out = (float*)d_out;

    float* packWx = (float*)d_ws;                                  // 4 MB
    float* packWh = packWx + (size_t)RNN_D * RNN_H;                // 4 MB
    unsigned* ctr = (unsigned*)((char*)d_ws + (size_t)8 * 1024 * 1024);

    // Pack B matrices into WMMA fragment order (64*256*32 threads each).
    rnn_pack_b_kernel<<<2048, 256, 0, stream>>>(Wx, packWx);
    rnn_pack_b_kernel<<<2048, 256, 0, stream>>>(Wh, packWh);

    // Zero the per-group barrier counters (graph-capture-safe memset node).
    hipMemsetAsync(ctr, 0, 8 * sizeof(unsigned), stream);

    // Phase 1: input projection into d_out.
    dim3 g1(4096, 8);
    rnn_xproj_kernel<<<g1, 256, 0, stream>>>(x, packWx, b, out);

    // Phase 2: sequential recurrence, in place over d_out.
    rnn_recur_kernel<<<128, 128, 0, stream>>>(h0, packWh, out, ctr);
}